// GeneralizedAdvantageEstimation_10153302688091
// MI455X (gfx1250) — compile-verified
//
#include <hip/hip_runtime.h>

// ---------------------------------------------------------------------------
// GAE (reverse-time linear recurrence) as a 3-pass blocked affine scan.
//   pass1: per chunk c, per column: P_c (partial with carry=0), A_c (prod of a_t)
//   pass2: serial scan over chunks: Gin[c] = carry entering chunk c from above
//   pass3: re-scan each chunk seeded with Gin, emit adv and ret = adv + v
// Memory-bound (~0.3 flop/byte): fp32 throughout, b128 coalesced traffic,
// async global->LDS double-buffered prefetch (CDNA5 ASYNCcnt path).
// ---------------------------------------------------------------------------

namespace {

constexpr float kGamma = 0.99f;
constexpr float kLam   = 0.95f;
constexpr float kGD    = kGamma * kLam;

constexpr int kT  = 4096;        // reference T (fixed in reference module)
constexpr int kTC = 32;          // timesteps per chunk
constexpr int kC  = kT / kTC;    // 128 chunks
constexpr int kBT = 256;         // threads per block (8 waves, wave32)

#ifndef __has_builtin
#define __has_builtin(x) 0
#endif

#if defined(__HIP_DEVICE_COMPILE__) && \
    __has_builtin(__builtin_amdgcn_global_load_async_to_lds_b128)
#define GAE_USE_ASYNC 1
#else
#define GAE_USE_ASYNC 0
#endif

#if GAE_USE_ASYNC
#if __has_builtin(__builtin_amdgcn_s_wait_asynccnt)
#define GAE_WAIT_ASYNC(n) __builtin_amdgcn_s_wait_asynccnt(n)
#else
#define GAE_WAIT_ASYNC(n) asm volatile("s_wait_asynccnt %0" ::"i"(n) : "memory")
#endif
// Builtin prototype (from hipcc diagnostic): param0 = int4 (vector_size 16)
// pointer in global AS; param1 = int4 pointer in LDS AS. AS3 pointers are
// 32 bits == LDS byte offset; generic LDS pointers carry the offset in their
// low 32 bits, so truncation through an integer is exact.
typedef int gae_v4i __attribute__((__vector_size__(4 * sizeof(int))));
#define GAE_AS1(p) \
  ((__attribute__((address_space(1))) gae_v4i*)(unsigned long long)(p))
#define GAE_AS3(p) \
  ((__attribute__((address_space(3))) gae_v4i*)(unsigned int)(unsigned long long)(p))
#endif

template <int FINAL>
__global__ __launch_bounds__(kBT) void gae_chunk_kernel(
    const float4* __restrict__ rewards, const float4* __restrict__ values,
    const float4* __restrict__ next_values, const float4* __restrict__ next_dones,
    float4* __restrict__ Pp, float4* __restrict__ Ap,
    const float4* __restrict__ Gin,
    float4* __restrict__ adv, float4* __restrict__ ret, int N4) {
  const int c = blockIdx.x;                       // chunk index, time-major
  const int j = blockIdx.y * kBT + threadIdx.x;   // float4 column index
  if (j >= N4) return;
  const int tTop = c * kTC + (kTC - 1);
  const int tid  = threadIdx.x;

  float4 g  = make_float4(0.f, 0.f, 0.f, 0.f);
  float4 Ac = make_float4(1.f, 1.f, 1.f, 1.f);
  if (FINAL) g = Gin[(size_t)c * N4 + j];

#if GAE_USE_ASYNC
  // 2 stages x 4 arrays x 256 lanes x 16B = 32 KB LDS (of 320 KB/WGP).
  __shared__ float4 buf[2][4][kBT];
  auto issue = [&](int t, int s) {
    const size_t base = (size_t)t * N4 + j;
    __builtin_amdgcn_global_load_async_to_lds_b128(GAE_AS1(rewards + base),
                                                   GAE_AS3(&buf[s][0][tid]), 0, 0);
    __builtin_amdgcn_global_load_async_to_lds_b128(GAE_AS1(values + base),
                                                   GAE_AS3(&buf[s][1][tid]), 0, 0);
    __builtin_amdgcn_global_load_async_to_lds_b128(GAE_AS1(next_values + base),
                                                   GAE_AS3(&buf[s][2][tid]), 0, 0);
    __builtin_amdgcn_global_load_async_to_lds_b128(GAE_AS1(next_dones + base),
                                                   GAE_AS3(&buf[s][3][tid]), 0, 0);
  };
#endif

  auto step = [&](int t, int s) {
    float4 r, v, nv, nd;
#if GAE_USE_ASYNC
    r  = buf[s][0][tid];
    v  = buf[s][1][tid];
    nv = buf[s][2][tid];
    nd = buf[s][3][tid];
#else
    (void)s;
    const size_t base = (size_t)t * N4 + j;
    r  = rewards[base];
    v  = values[base];
    nv = next_values[base];
    nd = next_dones[base];
#endif
#define GAE_C(X)                                                        \
  {                                                                     \
    float nnd   = 1.0f - nd.X;                                          \
    float a     = kGD * nnd;                                            \
    float delta = __builtin_fmaf(kGamma * nv.X, nnd, r.X) - v.X;        \
    g.X = __builtin_fmaf(a, g.X, delta);                                \
    if (!FINAL) Ac.X *= a;                                              \
  }
    GAE_C(x) GAE_C(y) GAE_C(z) GAE_C(w)
#undef GAE_C
    if (FINAL) {
      const size_t base2 = (size_t)t * N4 + j;
      adv[base2] = g;
      ret[base2] = make_float4(g.x + v.x, g.y + v.y, g.z + v.z, g.w + v.w);
    }
  };

#if GAE_USE_ASYNC
  issue(tTop, 0);
  for (int i = 0; i < kTC - 1; ++i) {
    issue(tTop - (i + 1), (i + 1) & 1);   // prefetch next row (4 async b128)
    GAE_WAIT_ASYNC(4);                    // in-order retire: current stage done
    asm volatile("" ::: "memory");
    step(tTop - i, i & 1);
  }
  GAE_WAIT_ASYNC(0);
  asm volatile("" ::: "memory");
  step(tTop - (kTC - 1), (kTC - 1) & 1);
#else
#pragma unroll 4
  for (int i = 0; i < kTC; ++i) step(tTop - i, 0);
#endif

  if (!FINAL) {
    const size_t o = (size_t)c * N4 + j;
    Pp[o] = g;
    Ap[o] = Ac;
  }
}

__global__ __launch_bounds__(256) void gae_carry_kernel(
    const float4* __restrict__ Pp, const float4* __restrict__ Ap,
    float4* __restrict__ Gin, int N4, int C) {
  const int j = blockIdx.x * 256 + threadIdx.x;
  if (j >= N4) return;
  float4 carry = make_float4(0.f, 0.f, 0.f, 0.f);
  for (int c = C - 1; c >= 0; --c) {
    const size_t o = (size_t)c * N4 + j;
    Gin[o] = carry;                       // carry entering chunk c from above
    const float4 p = Pp[o];
    const float4 a = Ap[o];
    carry.x = __builtin_fmaf(a.x, carry.x, p.x);
    carry.y = __builtin_fmaf(a.y, carry.y, p.y);
    carry.z = __builtin_fmaf(a.z, carry.z, p.z);
    carry.w = __builtin_fmaf(a.w, carry.w, p.w);
  }
}

}  // namespace

extern "C" void kernel_launch(void* const* d_in, const int* in_sizes, int n_in,
                              void* d_out, int out_size, void* d_ws, size_t ws_size,
                              hipStream_t stream) {
  (void)n_in; (void)out_size; (void)ws_size;
  const int T  = kT;
  const int N  = in_sizes[0] / T;   // 2048
  const int N4 = N / 4;
  const int C  = kC;

  const float4* rewards     = (const float4*)d_in[0];
  const float4* values      = (const float4*)d_in[1];
  const float4* next_values = (const float4*)d_in[2];
  const float4* next_dones  = (const float4*)d_in[3];

  float*  ws  = (float*)d_ws;                       // 3*C*N floats = 3 MB
  float4* Pp  = (float4*)(ws);
  float4* Ap  = (float4*)(ws + (size_t)C * N);
  float4* Gin = (float4*)(ws + 2 * (size_t)C * N);

  float4* adv = (float4*)d_out;
  float4* ret = (float4*)((float*)d_out + (size_t)T * N);

  dim3 grid1(C, (N4 + kBT - 1) / kBT);
  gae_chunk_kernel<0><<<grid1, kBT, 0, stream>>>(
      rewards, values, next_values, next_dones, Pp, Ap, nullptr, nullptr, nullptr, N4);
  gae_carry_kernel<<<(N4 + 255) / 256, 256, 0, stream>>>(Pp, Ap, Gin, N4, C);
  gae_chunk_kernel<1><<<grid1, kBT, 0, stream>>>(
      rewards, values, next_values, next_dones, nullptr, nullptr, Gin, adv, ret, N4);
}